// SemanticGraphFusion_73529840107837
// MI455X (gfx1250) — compile-verified
//
#include <hip/hip_runtime.h>

// MI455X / gfx1250, wave32. All matrix math via V_WMMA_F32_16X16X4_F32 (fp32,
// matching reference numerics). B matrices staged in LDS in a row-pair-
// interleaved layout so each WMMA B fragment is one ds_load_b64.

typedef __attribute__((ext_vector_type(2))) float v2f;
typedef __attribute__((ext_vector_type(4))) float v4f;
typedef __attribute__((ext_vector_type(8))) float v8f;

#define NG    120            // NUM_GRAPHS
#define GP    128            // padded graph count
#define GD    625            // GRAPH_DIM^2
#define HID   64             // HIDDEN
#define RPB   128            // sample rows per block
#define CHUNK 160            // flat-matrix column chunk staged in LDS
#define NEGINF (-3.0e38f)

// ---------------------------------------------------------------------------
// Kernel 1: Q = flat@Wq^T + bq ; K = flat@Wk^T + bk    (tiny: 120x64 each)
// ---------------------------------------------------------------------------
__global__ __launch_bounds__(128) void qk_kernel(
    const float* __restrict__ graphs,
    const float* __restrict__ Wq, const float* __restrict__ bq,
    const float* __restrict__ Wk, const float* __restrict__ bk,
    float* __restrict__ qbuf, float* __restrict__ kbuf)
{
  __shared__ float flat[GD];
  const int g = blockIdx.x;
  for (int d = threadIdx.x; d < GD; d += blockDim.x) flat[d] = graphs[g * GD + d];
  __syncthreads();
  const int t = threadIdx.x;
  const int h = t & 63;
  const float* W = (t < HID) ? Wq : Wk;
  const float* b = (t < HID) ? bq : bk;
  const float* wrow = W + (size_t)h * GD;
  float acc = b[h];
  for (int d = 0; d < GD; ++d) acc = fmaf(flat[d], wrow[d], acc);
  if (t < HID) qbuf[g * HID + h] = acc;
  else         kbuf[g * HID + h] = acc;
}

// ---------------------------------------------------------------------------
// Kernel 2: scoresP = scores^T in row-pair-interleaved layout:
//   element (k=j, col=i) stored at scoresP[(j>>1)*(2*GP) + i*2 + (j&1)]
// so B fragment rows (k,k+1) at one column are 8 contiguous bytes.
// grid = 128 (i), block = 128 (j); zeros outside the 120x120 region.
// ---------------------------------------------------------------------------
__global__ __launch_bounds__(128) void scores_kernel(
    const float* __restrict__ qbuf, const float* __restrict__ kbuf,
    float* __restrict__ scoresP)
{
  __shared__ float qi[HID];
  const int i = blockIdx.x;
  const int j = threadIdx.x;
  if (i < NG && j < HID) qi[j] = qbuf[i * HID + j];
  __syncthreads();
  float v = 0.0f;
  if (i < NG && j < NG) {
    const float* kr = kbuf + (size_t)j * HID;
    for (int h = 0; h < HID; ++h) v = fmaf(qi[h], kr[h], v);
  }
  scoresP[(j >> 1) * (2 * GP) + i * 2 + (j & 1)] = v;
}

// ---------------------------------------------------------------------------
// Kernel 3 (fused): mask -> means (WMMA) -> masked softmax -> attn@flat (WMMA)
// grid = N/128, block = 256 (8 waves, 16 sample rows per wave)
// Dynamic LDS layout:
//   [0,      65536)  smA    : scoresP[128][128] (paired), later attn[128][128]
//                             (row-major) -- aliased, barrier-separated
//   [65536, 142336)  smF    : flat chunk, paired: [(g>>1)][c*2 + (g&1)]
//   [142336,144384)  smMask : 128 rows x 4 u32 mask words
//   [144384,144896)  smCnt  : 128 floats (selected count per row)
// ---------------------------------------------------------------------------
__global__ __launch_bounds__(256) void fused_kernel(
    const float* __restrict__ logits, const float* __restrict__ graphs,
    const float* __restrict__ scoresP, float* __restrict__ out)
{
  extern __shared__ char smem[];
  float*    smA    = (float*)smem;
  float*    smF    = (float*)(smem + 65536);
  unsigned* smMask = (unsigned*)(smem + 65536 + 76800);
  float*    smCnt  = (float*)(smem + 65536 + 76800 + 2048);

  const int tid   = threadIdx.x;
  const int wid   = tid >> 5;
  const int lane  = tid & 31;
  const int lrow  = lane & 15;   // row-in-tile (A layout) / col-in-tile (B/D)
  const int hv    = lane >> 4;   // half-wave
  const int nbase = blockIdx.x * RPB;

  // ---- stage A: cooperative load of paired scores into LDS ----
  {
    const v4f* src = (const v4f*)scoresP;
    v4f* dst = (v4f*)smA;
    for (int i = tid; i < (GP * GP) / 4; i += 256) dst[i] = src[i];
  }

  // ---- stage B: per-row max / argmax / mask words (lane owns one row,
  //      halves split the 120 columns as 64 + 56) ----
  const int r = wid * 16 + lrow;                 // block-local row this lane owns
  const float* lp = logits + (size_t)(nbase + r) * NG;
  const int j0 = hv * 64;
  const int j1 = hv ? NG : 64;
  float mx = NEGINF;
  int arg = j0;
  for (int j = j0; j < j1; ++j) {
    float x = lp[j];
    if (x > mx) { mx = x; arg = j; }
  }
  {
    float omx = __shfl_xor(mx, 16, 32);
    int   oarg = __shfl_xor(arg, 16, 32);
    if (omx > mx || (omx == mx && oarg < arg)) { mx = omx; arg = oarg; }
  }
  {
    const bool useThr = (mx > 0.0f);            // empty-mask fallback <=> max <= 0
    const float thr = 0.5f * mx;
    unsigned w0 = 0, w1 = 0;
    for (int j = j0; j < j1; ++j) {
      float x = lp[j];
      bool m = useThr ? (x > thr) : (j == arg);
      int jj = j - j0;
      if (m) { if (jj < 32) w0 |= 1u << jj; else w1 |= 1u << (jj - 32); }
    }
    smMask[r * 4 + hv * 2 + 0] = w0;
    smMask[r * 4 + hv * 2 + 1] = w1;
  }
  __syncthreads();
  if (hv == 0) {
    unsigned a0 = smMask[r*4+0], a1 = smMask[r*4+1], a2 = smMask[r*4+2], a3 = smMask[r*4+3];
    smCnt[r] = (float)(__popc(a0) + __popc(a1) + __popc(a2) + __popc(a3));
  }
  __syncthreads();

  // ---- stage C: A-fragments = mask as 0/1 floats (16x120, K in steps of 4) ----
  v2f amf[30];
  {
    unsigned m0 = smMask[r*4+0], m1 = smMask[r*4+1], m2 = smMask[r*4+2], m3 = smMask[r*4+3];
#pragma unroll
    for (int s = 0; s < 30; ++s) {
      int k0 = 4 * s + 2 * hv;
      unsigned wa = (k0     < 32) ? m0 : (k0     < 64) ? m1 : (k0     < 96) ? m2 : m3;
      unsigned wb = (k0 + 1 < 32) ? m0 : (k0 + 1 < 64) ? m1 : (k0 + 1 < 96) ? m2 : m3;
      amf[s].x = (float)((wa >> (k0 & 31)) & 1u);
      amf[s].y = (float)((wb >> ((k0 + 1) & 31)) & 1u);
    }
  }

  // ---- stage D: means = mf @ scores^T via v_wmma_f32_16x16x4_f32.
  //      B fragment = one 64-bit LDS load from the paired layout. ----
  v8f dmat[8];
#pragma unroll
  for (int t = 0; t < 8; ++t) {
    v8f acc = {0,0,0,0,0,0,0,0};
#pragma unroll
    for (int s = 0; s < 30; ++s) {
      const int p = 2 * s + hv;                       // (k>>1), k = 4s+2hv
      const v2f b = *(const v2f*)&smA[p * (2 * GP) + (t * 16 + lrow) * 2];
      acc = __builtin_amdgcn_wmma_f32_16x16x4_f32(false, amf[s], false, b,
                                                  (short)0, acc, false, false);
    }
    dmat[t] = acc;
  }

  // ---- stage E: masked softmax on (means / cnt), in WMMA D layout ----
#pragma unroll
  for (int v = 0; v < 8; ++v) {
    const int myrow = wid * 16 + hv * 8 + v;
    const float ic = 1.0f / smCnt[myrow];
    unsigned m0 = smMask[myrow*4+0], m1 = smMask[myrow*4+1],
             m2 = smMask[myrow*4+2], m3 = smMask[myrow*4+3];
    float vals[8]; bool msk[8];
    float vmax = NEGINF;
#pragma unroll
    for (int t = 0; t < 8; ++t) {
      int col = t * 16 + lrow;
      unsigned w = (col < 32) ? m0 : (col < 64) ? m1 : (col < 96) ? m2 : m3;
      msk[t] = ((w >> (col & 31)) & 1u) != 0u;
      vals[t] = msk[t] ? dmat[t][v] * ic : NEGINF;
      vmax = fmaxf(vmax, vals[t]);
    }
#pragma unroll
    for (int off = 1; off < 16; off <<= 1)
      vmax = fmaxf(vmax, __shfl_xor(vmax, off, 32));
    float ev[8]; float ssum = 0.0f;
#pragma unroll
    for (int t = 0; t < 8; ++t) {
      ev[t] = msk[t] ? expf(vals[t] - vmax) : 0.0f;
      ssum += ev[t];
    }
#pragma unroll
    for (int off = 1; off < 16; off <<= 1)
      ssum += __shfl_xor(ssum, off, 32);
    const float inv = 1.0f / ssum;
#pragma unroll
    for (int t = 0; t < 8; ++t) dmat[t][v] = ev[t] * inv;
  }

  // ---- stage F: publish attn (row-major) into smA; all score reads done ----
  __syncthreads();
#pragma unroll
  for (int t = 0; t < 8; ++t)
#pragma unroll
    for (int v = 0; v < 8; ++v)
      smA[(wid * 16 + hv * 8 + v) * GP + t * 16 + lrow] = dmat[t][v];
  __syncthreads();

  // ---- stage G: preload stage-2 A fragments (attn rows; (k,k+1) contiguous
  //      and 8B-aligned in the row-major attn buffer -> b64 loads) ----
  v2f aat[30];
#pragma unroll
  for (int s = 0; s < 30; ++s) {
    const int k = 4 * s + 2 * hv;
    aat[s] = *(const v2f*)&smA[r * GP + k];
  }

  // ---- stage H: out = attn @ flat, 625 cols in 4 LDS chunks of 160.
  //      flat chunk staged paired so B fragments are single b64 loads. ----
  for (int ch = 0; ch < 4; ++ch) {
    const int c0 = ch * CHUNK;
    __syncthreads();                       // prior chunk fully consumed
    for (int idx = tid; idx < NG * CHUNK; idx += 256) {
      int g = idx / CHUNK;
      int c = idx - g * CHUNK;
      int col = c0 + c;
      float vv = (col < GD) ? graphs[(size_t)g * GD + col] : 0.0f;
      smF[(g >> 1) * (2 * CHUNK) + c * 2 + (g & 1)] = vv;
    }
    __syncthreads();
    for (int ct = 0; ct < CHUNK / 16; ++ct) {
      v8f acc = {0,0,0,0,0,0,0,0};
#pragma unroll
      for (int s = 0; s < 30; ++s) {
        const int p = 2 * s + hv;                     // (k>>1)
        const v2f b = *(const v2f*)&smF[p * (2 * CHUNK) + (ct * 16 + lrow) * 2];
        acc = __builtin_amdgcn_wmma_f32_16x16x4_f32(false, aat[s], false, b,
                                                    (short)0, acc, false, false);
      }
      const int col = c0 + ct * 16 + lrow;
      if (col < GD) {
        const size_t rbase = (size_t)(nbase + wid * 16 + hv * 8);
#pragma unroll
        for (int v = 0; v < 8; ++v)
          out[(rbase + v) * GD + col] = acc[v];
      }
    }
  }
}

// ---------------------------------------------------------------------------
extern "C" void kernel_launch(void* const* d_in, const int* in_sizes, int n_in,
                              void* d_out, int out_size, void* d_ws, size_t ws_size,
                              hipStream_t stream) {
  const float* logits = (const float*)d_in[0];
  const float* graphs = (const float*)d_in[1];
  const float* Wq     = (const float*)d_in[2];
  const float* bq     = (const float*)d_in[3];
  const float* Wk     = (const float*)d_in[4];
  const float* bk     = (const float*)d_in[5];
  float* out = (float*)d_out;

  float* ws      = (float*)d_ws;
  float* scoresP = ws;                   // 128*128 = 16384 floats (paired layout)
  float* qbuf    = ws + GP * GP;         // 120*64
  float* kbuf    = qbuf + NG * HID;      // 120*64

  const int N = in_sizes[0] / NG;        // 32768

  qk_kernel<<<NG, 128, 0, stream>>>(graphs, Wq, bq, Wk, bk, qbuf, kbuf);
  scores_kernel<<<GP, GP, 0, stream>>>(qbuf, kbuf, scoresP);

  const size_t shmem = 65536 + 76800 + 2048 + 512;   // 144896 B (<320KB/WGP)
  fused_kernel<<<N / RPB, 256, shmem, stream>>>(logits, graphs, scoresP, out);
}